// node_convnet_feat_22239340658707
// MI455X (gfx1250) — compile-verified
//
#include <hip/hip_runtime.h>

// Problem constants (from reference): B=16, V=128, H=256
#define BB 16
#define VV 128
#define HH 256

typedef float v2f __attribute__((ext_vector_type(2)));
typedef float v4f __attribute__((ext_vector_type(4)));
typedef float v8f __attribute__((ext_vector_type(8)));

// ---------------------------------------------------------------------------
// Kernel 1: Ux = x @ U_w^T + U_b ; Vx = x @ V_w^T + V_b   via V_WMMA_F32_16X16X4_F32
// One wave computes one 16x16 output tile for BOTH U and V (shared A fragment).
// Tiles: B * (V/16) * (H/16) = 16*8*16 = 2048 waves = 256 blocks x 8 waves.
// Ux -> ws[2MB..4MB), Vx -> ws[0..2MB)
// ---------------------------------------------------------------------------
__global__ __launch_bounds__(256)
void gemm_uv_wmma(const float* __restrict__ x,
                  const float* __restrict__ Uw, const float* __restrict__ Ub,
                  const float* __restrict__ Vw, const float* __restrict__ Vb,
                  float* __restrict__ UxOut, float* __restrict__ VxOut)
{
    const int lane = threadIdx.x & 31;
    const int wave = threadIdx.x >> 5;
    const int wid  = blockIdx.x * 8 + wave;   // 0..2047

    const int ot = wid & 15;          // o-tile: 16 tiles of 16 over H
    const int vt = (wid >> 4) & 7;    // v-tile: 8 tiles of 16 over V
    const int b  = wid >> 7;          // batch

    const int o0 = ot * 16;
    const int v0 = vt * 16;

    // A fragment (16x4 f32): lanes 0-15 -> M=lane, K pair (k0+0,k0+1);
    // lanes 16-31 -> M=lane-16, K pair (k0+2,k0+3).
    const int m    = lane & 15;
    const int koff = (lane >> 4) << 1;     // 0 or 2

    const float* __restrict__ xp = x  + ((size_t)b * VV + (v0 + m)) * HH + koff;
    // B fragment (4x16): B[k][n] = W[o0+n][k]; same lane pattern -> contiguous pair in h.
    const float* __restrict__ up = Uw + (size_t)(o0 + m) * HH + koff;
    const float* __restrict__ vp = Vw + (size_t)(o0 + m) * HH + koff;

    v8f cu = {};
    v8f cv = {};

    #pragma unroll 4
    for (int k0 = 0; k0 < HH; k0 += 4) {
        v2f a  = *(const v2f*)(xp + k0);
        v2f bu = *(const v2f*)(up + k0);
        v2f bv = *(const v2f*)(vp + k0);
        // (neg_a, A, neg_b, B, c_mod, C, reuse_a, reuse_b)
        cu = __builtin_amdgcn_wmma_f32_16x16x4_f32(false, a, false, bu, (short)0, cu, false, false);
        cv = __builtin_amdgcn_wmma_f32_16x16x4_f32(false, a, false, bv, (short)0, cv, false, false);
    }

    // C/D layout: VGPR r -> M = r + (lane>>4)*8 ; N = lane & 15
    const int n   = lane & 15;
    const float biasU = Ub[o0 + n];
    const float biasV = Vb[o0 + n];
    const int mhalf = (lane >> 4) * 8;

    #pragma unroll
    for (int r = 0; r < 8; ++r) {
        const int vrow = v0 + r + mhalf;
        const size_t idx = ((size_t)b * VV + vrow) * HH + o0 + n;
        UxOut[idx] = cu[r] + biasU;
        VxOut[idx] = cv[r] + biasV;
    }
}

// ---------------------------------------------------------------------------
// Kernel 2: out[b,i,h] = Ux[b,i,h] + sum_j gate[b,i,j,h] * Vx[b,j,h]
// Bandwidth bound: streams 268 MB of gate exactly once (non-temporal loads so
// the 2 MB Vx stays resident in the 192 MB L2 across its ~128x re-reads).
// Block: 256 threads = 4 i-rows x 64 h-quads. Grid: B * V/4 = 512 blocks.
// Pure write to d_out (replay-safe).
// ---------------------------------------------------------------------------
__global__ __launch_bounds__(256)
void agg_stream(const float* __restrict__ gate,
                const float* __restrict__ Vx,
                const float* __restrict__ Ux,
                float* __restrict__ out)
{
    const int hq = threadIdx.x & 63;       // h/4
    const int il = threadIdx.x >> 6;       // 0..3
    const int i  = (blockIdx.x & 31) * 4 + il;
    const int b  = blockIdx.x >> 5;
    const int h  = hq << 2;

    const v4f* __restrict__ gp = (const v4f*)(gate + (((size_t)(b * VV + i)) * VV) * HH + h);
    const v4f* __restrict__ vp = (const v4f*)(Vx + ((size_t)b * VV) * HH + h);

    v4f acc = {0.0f, 0.0f, 0.0f, 0.0f};

    #pragma unroll 4
    for (int j = 0; j < VV; ++j) {
        v4f g = __builtin_nontemporal_load(gp + (size_t)j * (HH / 4));  // streamed once: NT
        v4f v = vp[(size_t)j * (HH / 4)];                               // L2-resident: RT
        acc += g * v;
    }

    const size_t oidx = ((size_t)(b * VV + i)) * HH + h;
    v4f u = *(const v4f*)(Ux + oidx);
    *(v4f*)(out + oidx) = u + acc;
}

// ---------------------------------------------------------------------------
extern "C" void kernel_launch(void* const* d_in, const int* in_sizes, int n_in,
                              void* d_out, int out_size, void* d_ws, size_t ws_size,
                              hipStream_t stream)
{
    const float* x    = (const float*)d_in[0];   // [B,V,H]
    const float* gate = (const float*)d_in[1];   // [B,V,V,H]
    const float* Uw   = (const float*)d_in[2];   // [H,H]
    const float* Ub   = (const float*)d_in[3];   // [H]
    const float* Vw   = (const float*)d_in[4];   // [H,H]
    const float* Vb   = (const float*)d_in[5];   // [H]
    float* out = (float*)d_out;

    // Workspace layout: [0 .. BVH) = Vx, [BVH .. 2*BVH) = Ux  (floats)
    const size_t BVH = (size_t)BB * VV * HH;     // 524288 floats = 2 MB
    float* VxWs = (float*)d_ws;
    float* UxWs = VxWs + BVH;

    // GEMMs: 2048 waves -> 256 blocks of 8 waves
    gemm_uv_wmma<<<256, 256, 0, stream>>>(x, Uw, Ub, Vw, Vb, UxWs, VxWs);

    // Aggregation + epilogue: 512 blocks of 256 threads
    agg_stream<<<BB * (VV / 4), 256, 0, stream>>>(gate, VxWs, UxWs, out);
}